// SAGE_49701361549349
// MI455X (gfx1250) — compile-verified
//
#include <hip/hip_runtime.h>

#define N_NODES 50000
#define N_EDGES 800000
#define HDIM    128
#define N_GRAPHS 512
#define KPS     288   // LDS words per interleaved k-pair row (padded: 288%64==32 -> conflict-free halves)

typedef float v8f __attribute__((ext_vector_type(8)));
typedef float v2f __attribute__((ext_vector_type(2)));

// ---------------------------------------------------------------- utilities
__global__ __launch_bounds__(256) void sage_zero_f32(float* __restrict__ p, size_t n) {
  size_t i = (size_t)blockIdx.x * 256 + threadIdx.x;
  size_t stride = (size_t)gridDim.x * 256;
  for (; i < n; i += stride) p[i] = 0.0f;
}

__global__ __launch_bounds__(256) void sage_deg_count(const int* __restrict__ dst,
                                                      float* __restrict__ deg) {
  int e = blockIdx.x * 256 + threadIdx.x;
  if (e < N_EDGES) atomicAdd(&deg[dst[e]], 1.0f);
}

__global__ __launch_bounds__(256) void sage_deg_invert(float* __restrict__ deg) {
  int i = blockIdx.x * 256 + threadIdx.x;
  if (i < N_NODES) deg[i] = 1.0f / fmaxf(deg[i], 1.0f);
}

// x[i,:] = z_emb[z[i],:]  (one wave per node row, float4 lanes)
__global__ __launch_bounds__(256) void sage_gather_emb(const int* __restrict__ z,
                                                       const float* __restrict__ z_emb,
                                                       float* __restrict__ x) {
  int t = blockIdx.x * 256 + threadIdx.x;
  int i = t >> 5;
  int c = (t & 31) << 2;
  if (i < N_NODES) {
    int r = z[i];
    float4 v = *(const float4*)(z_emb + (size_t)r * HDIM + c);
    *(float4*)(x + (size_t)i * HDIM + c) = v;
  }
}

// agg[dst[e],:] += x[src[e],:]  (one wave per edge, float4 lanes, f32 atomics)
__global__ __launch_bounds__(256) void sage_scatter_add(const float* __restrict__ x,
                                                        const int* __restrict__ src,
                                                        const int* __restrict__ dst,
                                                        float* __restrict__ agg) {
  long long t = (long long)blockIdx.x * 256 + threadIdx.x;
  int e = (int)(t >> 5);
  int c = ((int)t & 31) << 2;
  if (e < N_EDGES) {
    int s = src[e], d = dst[e];
    float4 v = *(const float4*)(x + (size_t)s * HDIM + c);
    float* ap = agg + (size_t)d * HDIM + c;
    atomicAdd(ap + 0, v.x);
    atomicAdd(ap + 1, v.y);
    atomicAdd(ap + 2, v.z);
    atomicAdd(ap + 3, v.w);
  }
}

// g[batch[i],:] += x[i,:]
__global__ __launch_bounds__(256) void sage_pool_add(const float* __restrict__ x,
                                                     const int* __restrict__ batch,
                                                     float* __restrict__ g) {
  int t = blockIdx.x * 256 + threadIdx.x;
  int i = t >> 5;
  int c = (t & 31) << 2;
  if (i < N_NODES) {
    int b = batch[i];
    float4 v = *(const float4*)(x + (size_t)i * HDIM + c);
    float* gp = g + (size_t)b * HDIM + c;
    atomicAdd(gp + 0, v.x);
    atomicAdd(gp + 1, v.y);
    atomicAdd(gp + 2, v.z);
    atomicAdd(gp + 3, v.w);
  }
}

// ------------------------------------------------------ fused SAGE GEMM (WMMA)
// out = act( (Aagg * dinv) @ Wl + bl  [+ Ax @ Wr] )
// block = 256 threads = 8 waves; wave -> 16 rows x 128 cols of output.
// Weights staged through LDS in K-chunks of 32 rows, K-PAIR INTERLEAVED:
//   lds[p*KPS + n*2 + {0,1}] = { W[kc+2p][n], W[kc+2p+1][n] }
// so a lane's B fragment is ONE aligned ds_load_b64 straight into the WMMA
// operand pair (no repacking movs), and the two half-waves (pair p vs p+1)
// read disjoint LDS bank halves thanks to the 288-word stride.
__global__ __launch_bounds__(256) void sage_gemm_wmma(
    const float* __restrict__ Aagg, const float* __restrict__ deg_inv,
    const float* __restrict__ Ax,
    const float* __restrict__ Wl, const float* __restrict__ bl,
    const float* __restrict__ Wr,
    float* __restrict__ out, int nrows, int do_relu, int has_root) {
  __shared__ __align__(16) float lds_wl[16 * KPS];
  __shared__ __align__(16) float lds_wr[16 * KPS];
  __shared__ float lds_b[HDIM];

  const int tid = threadIdx.x;
  if (tid < HDIM) lds_b[tid] = bl[tid];

  const int wave  = tid >> 5;
  const int lane  = tid & 31;
  const int khalf = lane >> 4;         // 0 or 1: which K-pair of the 4-wide K step
  const int klo   = khalf << 1;        // 0 or 2
  const int nl    = lane & 15;         // B/C/D fragment N index
  const int mrow  = lane & 15;         // A fragment M index

  const int row_base = blockIdx.x * 128 + wave * 16;
  int row  = row_base + mrow;
  int rowc = row < nrows ? row : nrows - 1;       // clamp: WMMA needs EXEC all-1s
  const float dinv = deg_inv ? deg_inv[rowc] : 1.0f;
  const float* aggp = Aagg + (size_t)rowc * HDIM;
  const float* xp   = has_root ? (Ax + (size_t)rowc * HDIM) : Aagg;

  v8f acc[8] = {};

  for (int kc = 0; kc < HDIM; kc += 32) {
    __syncthreads();
    // stage 32 K-rows of Wl (and Wr), interleaved by K-pair
    for (int task = tid; task < 512; task += 256) {
      int rp = task >> 5;          // k-pair 0..15
      int q  = task & 31;          // float4 column 0..31
      {
        const float* gp = Wl + (size_t)(kc + 2 * rp) * HDIM + q * 4;
        float4 e = *(const float4*)gp;
        float4 o = *(const float4*)(gp + HDIM);
        float* dp = lds_wl + rp * KPS + q * 8;
        ((float4*)dp)[0] = make_float4(e.x, o.x, e.y, o.y);
        ((float4*)dp)[1] = make_float4(e.z, o.z, e.w, o.w);
      }
      if (has_root) {
        const float* gp = Wr + (size_t)(kc + 2 * rp) * HDIM + q * 4;
        float4 e = *(const float4*)gp;
        float4 o = *(const float4*)(gp + HDIM);
        float* dp = lds_wr + rp * KPS + q * 8;
        ((float4*)dp)[0] = make_float4(e.x, o.x, e.y, o.y);
        ((float4*)dp)[1] = make_float4(e.z, o.z, e.w, o.w);
      }
    }
    __syncthreads();

    if (has_root) {
#pragma unroll
      for (int kk = 0; kk < 32; kk += 4) {
        int k  = kc + kk;
        int kp = (kk >> 1) + khalf;                 // interleaved pair index
        const float* bwl = lds_wl + kp * KPS + nl * 2;
        const float* bwr = lds_wr + kp * KPS + nl * 2;
        v2f a0 = *(const v2f*)(aggp + k + klo);
        a0.x *= dinv; a0.y *= dinv;
        v2f a1 = *(const v2f*)(xp + k + klo);
#pragma unroll
        for (int t = 0; t < 8; ++t) {
          v2f b0 = *(const v2f*)(bwl + t * 32);
          acc[t] = __builtin_amdgcn_wmma_f32_16x16x4_f32(
              false, a0, false, b0, (short)0, acc[t], false, false);
          v2f b1 = *(const v2f*)(bwr + t * 32);
          acc[t] = __builtin_amdgcn_wmma_f32_16x16x4_f32(
              false, a1, false, b1, (short)0, acc[t], false, false);
        }
      }
    } else {
#pragma unroll
      for (int kk = 0; kk < 32; kk += 4) {
        int k  = kc + kk;
        int kp = (kk >> 1) + khalf;
        const float* bwl = lds_wl + kp * KPS + nl * 2;
        v2f a0 = *(const v2f*)(aggp + k + klo);
        a0.x *= dinv; a0.y *= dinv;
#pragma unroll
        for (int t = 0; t < 8; ++t) {
          v2f b0 = *(const v2f*)(bwl + t * 32);
          acc[t] = __builtin_amdgcn_wmma_f32_16x16x4_f32(
              false, a0, false, b0, (short)0, acc[t], false, false);
        }
      }
    }
  }

  const int mhi = khalf << 3;  // C/D: lanes 16-31 hold M = vgpr + 8
#pragma unroll
  for (int t = 0; t < 8; ++t) {
    float bias = lds_b[t * 16 + nl];
#pragma unroll
    for (int r = 0; r < 8; ++r) {
      int orow = row_base + mhi + r;
      if (orow < nrows) {
        float v = acc[t][r] + bias;
        if (do_relu) v = fmaxf(v, 0.0f);
        out[(size_t)orow * HDIM + t * 16 + nl] = v;
      }
    }
  }
}

// out[g] = dot(h[g,:], W2) + b2
__global__ __launch_bounds__(128) void sage_mlp_out(const float* __restrict__ h,
                                                    const float* __restrict__ W2,
                                                    const float* __restrict__ b2,
                                                    float* __restrict__ out) {
  __shared__ float red[128];
  int g = blockIdx.x, j = threadIdx.x;
  red[j] = h[(size_t)g * HDIM + j] * W2[j];
  __syncthreads();
  for (int s = 64; s > 0; s >>= 1) {
    if (j < s) red[j] += red[j + s];
    __syncthreads();
  }
  if (j == 0) out[g] = red[0] + b2[0];
}

// ---------------------------------------------------------------- launcher
extern "C" void kernel_launch(void* const* d_in, const int* in_sizes, int n_in,
                              void* d_out, int out_size, void* d_ws, size_t ws_size,
                              hipStream_t stream) {
  (void)in_sizes; (void)n_in; (void)out_size; (void)ws_size;
  const int*   z     = (const int*)d_in[0];
  const int*   eidx  = (const int*)d_in[1];
  const int*   batch = (const int*)d_in[2];
  const float* z_emb = (const float*)d_in[3];
  const float* Wl[3] = {(const float*)d_in[4], (const float*)d_in[7],  (const float*)d_in[10]};
  const float* bl[3] = {(const float*)d_in[5], (const float*)d_in[8],  (const float*)d_in[11]};
  const float* Wr[3] = {(const float*)d_in[6], (const float*)d_in[9],  (const float*)d_in[12]};
  const float* W1 = (const float*)d_in[13];
  const float* b1 = (const float*)d_in[14];
  const float* W2 = (const float*)d_in[15];
  const float* b2 = (const float*)d_in[16];

  const int* src = eidx;             // edge_index[0]
  const int* dst = eidx + N_EDGES;   // edge_index[1]

  float* ws   = (float*)d_ws;
  float* deg  = ws;                                   // N
  float* x0   = deg + N_NODES;                        // N*H
  float* x1   = x0 + (size_t)N_NODES * HDIM;          // N*H
  float* agg  = x1 + (size_t)N_NODES * HDIM;          // N*H
  float* gbuf = agg + (size_t)N_NODES * HDIM;         // G*H
  float* hbuf = gbuf + (size_t)N_GRAPHS * HDIM;       // G*H

  const int ZGRID = 2048;
  // degree + inverse
  sage_zero_f32<<<ZGRID, 256, 0, stream>>>(deg, (size_t)N_NODES);
  sage_deg_count<<<(N_EDGES + 255) / 256, 256, 0, stream>>>(dst, deg);
  sage_deg_invert<<<(N_NODES + 255) / 256, 256, 0, stream>>>(deg);

  // embedding gather
  sage_gather_emb<<<(N_NODES * 32 + 255) / 256, 256, 0, stream>>>(z, z_emb, x0);

  const int scatter_blocks = (int)(((long long)N_EDGES * 32 + 255) / 256);
  const int gemm_blocks = (N_NODES + 127) / 128;

  float* xin = x0;
  float* xout = x1;
  for (int l = 0; l < 3; ++l) {
    sage_zero_f32<<<ZGRID, 256, 0, stream>>>(agg, (size_t)N_NODES * HDIM);
    sage_scatter_add<<<scatter_blocks, 256, 0, stream>>>(xin, src, dst, agg);
    sage_gemm_wmma<<<gemm_blocks, 256, 0, stream>>>(
        agg, deg, xin, Wl[l], bl[l], Wr[l], xout, N_NODES,
        /*do_relu=*/(l < 2) ? 1 : 0, /*has_root=*/1);
    float* tmp = xin; xin = xout; xout = tmp;
  }
  // xin now holds the final node features

  // global add pool
  sage_zero_f32<<<ZGRID, 256, 0, stream>>>(gbuf, (size_t)N_GRAPHS * HDIM);
  sage_pool_add<<<(N_NODES * 32 + 255) / 256, 256, 0, stream>>>(xin, batch, gbuf);

  // h = relu(g @ W1 + b1) via the same WMMA kernel (no root term, no deg scale)
  sage_gemm_wmma<<<(N_GRAPHS + 127) / 128, 256, 0, stream>>>(
      gbuf, nullptr, nullptr, W1, b1, nullptr, hbuf, N_GRAPHS,
      /*do_relu=*/1, /*has_root=*/0);

  // out = h @ W2 + b2
  sage_mlp_out<<<N_GRAPHS, 128, 0, stream>>>(hbuf, W2, b2, (float*)d_out);
}